// Attention_31396210933853
// MI455X (gfx1250) — compile-verified
//
#include <hip/hip_runtime.h>
#include <hip/hip_bf16.h>

#define B_  32
#define S_  4096
#define D_  512
#define NS  128            // s-rows per block
#define NCH (S_/NS)        // 32 chunks per batch row
#define EPSV 1e-7f

typedef __attribute__((ext_vector_type(2))) float v2f;
typedef __attribute__((ext_vector_type(4))) float v4f;
typedef __attribute__((ext_vector_type(8))) float v8f;

// Stage 1: per (b, s-chunk) block. Phase A: w[s] = exp(tanh(x.k + bias)) * mask.
// Phase C: WMMA pooling  part[ch][b][d] = sum_s w[s]*x[b,s,d]  (unnormalized).
__global__ __launch_bounds__(256) void attn_pool_stage1(
    const float* __restrict__ x, const float* __restrict__ kern,
    const float* __restrict__ bias, const unsigned char* __restrict__ mask,
    float* __restrict__ part, float* __restrict__ wsums)
{
    __shared__ float sk[D_];   // kernel vector (2 KB)
    __shared__ float sw[NS];   // weights for this chunk
    __shared__ float sred[8];  // per-wave weight sums

    const int tid  = threadIdx.x;
    const int lane = tid & 31;
    const int wv   = tid >> 5;          // 0..7
    const int b    = blockIdx.x;        // 0..31
    const int ch   = blockIdx.y;        // 0..NCH-1
    const int s0   = ch * NS;

    sk[tid]       = kern[tid];
    sk[tid + 256] = kern[tid + 256];

    const float* __restrict__ xb = x + (size_t)b * S_ * D_;
    __syncthreads();

    // ---------------- Phase A: eij -> weights (16 rows per wave) -------------
    float wsum_l = 0.f;
    for (int i = 0; i < 16; ++i) {
        const int rl = wv * 16 + i;
        const int s  = s0 + rl;
        const float* __restrict__ xr = xb + (size_t)s * D_;
        float e = 0.f;
        #pragma unroll
        for (int j = 0; j < 4; ++j) {
            const int d = j * 128 + lane * 4;      // coalesced 512B/wave/iter
            v4f xv = *(const v4f*)(xr + d);
            v4f kv = *(const v4f*)(sk + d);
            e += xv.x*kv.x + xv.y*kv.y + xv.z*kv.z + xv.w*kv.w;
        }
        #pragma unroll
        for (int off = 16; off > 0; off >>= 1)     // wave32 reduction
            e += __shfl_xor(e, off, 32);
        float w = __expf(tanhf(e + bias[s]));
        w *= (mask[(size_t)b * S_ + s] != 0) ? 1.f : 0.f;
        if (lane == 0) sw[rl] = w;
        wsum_l += w;                                // identical in all lanes
    }
    if (lane == 0) sred[wv] = wsum_l;
    __syncthreads();
    if (tid == 0) {
        float t = 0.f;
        #pragma unroll
        for (int i = 0; i < 8; ++i) t += sred[i];
        wsums[ch * B_ + b] = t;
    }

    // ---------------- Phase C: WMMA weighted pooling -------------------------
    // D(16x16) = A(16x4) * B(4x16) + C.  A = w[ss..ss+3] broadcast to all 16
    // rows; B = x tile (4 s-rows x 16 d-cols, row-major -> coalesced loads).
    // A 16x4 layout: lanes 0-15 hold K=0 (V0), K=1 (V1); lanes 16-31 K=2,3.
    // B  4x16 layout: lanes 0-15 hold K=0 (V0), K=1 (V1); lanes 16-31 K=2,3,
    // with N = lane%16.  All 16 D rows equal the pooled partial.
    const int d0   = wv * 64;                // this wave's 64 d-columns
    const int rsel = (lane >> 4) << 1;       // 0 for lanes 0-15, 2 for 16-31
    const int nn   = lane & 15;              // N column within 16-wide chunk
    v8f c0 = {}, c1 = {}, c2 = {}, c3 = {};

    for (int ss = 0; ss < NS; ss += 4) {
        v2f a;
        a.x = sw[ss + rsel];                 // broadcast LDS reads
        a.y = sw[ss + rsel + 1];
        const float* __restrict__ xp =
            xb + (size_t)(s0 + ss + rsel) * D_ + nn;
        v2f b0, b1, b2, b3;
        b0.x = xp[d0 +  0]; b0.y = xp[d0 +  0 + D_];
        b1.x = xp[d0 + 16]; b1.y = xp[d0 + 16 + D_];
        b2.x = xp[d0 + 32]; b2.y = xp[d0 + 32 + D_];
        b3.x = xp[d0 + 48]; b3.y = xp[d0 + 48 + D_];
        c0 = __builtin_amdgcn_wmma_f32_16x16x4_f32(false, a, false, b0, (short)0, c0, false, false);
        c1 = __builtin_amdgcn_wmma_f32_16x16x4_f32(false, a, false, b1, (short)0, c1, false, false);
        c2 = __builtin_amdgcn_wmma_f32_16x16x4_f32(false, a, false, b2, (short)0, c2, false, false);
        c3 = __builtin_amdgcn_wmma_f32_16x16x4_f32(false, a, false, b3, (short)0, c3, false, false);
    }

    // D VGPR0, lanes 0-15: M=0 row, N=lane -> pooled value for column d0+nn
    float* __restrict__ pp = part + ((size_t)ch * B_ + b) * D_ + d0;
    if (lane < 16) {
        pp[nn +  0] = c0[0];
        pp[nn + 16] = c1[0];
        pp[nn + 32] = c2[0];
        pp[nn + 48] = c3[0];
    }
}

// Stage 2: reduce 32 chunk partials and normalize by (sum_s w + eps).
__global__ __launch_bounds__(512) void attn_finalize(
    const float* __restrict__ part, const float* __restrict__ wsums,
    float* __restrict__ out)
{
    const int b = blockIdx.x;
    const int d = threadIdx.x;          // 512 threads == D
    float s = 0.f;
    #pragma unroll 4
    for (int c = 0; c < NCH; ++c) s += part[((size_t)c * B_ + b) * D_ + d];
    float ws = 0.f;
    #pragma unroll
    for (int c = 0; c < NCH; ++c) ws += wsums[c * B_ + b];
    out[b * D_ + d] = s / (ws + EPSV);
}

extern "C" void kernel_launch(void* const* d_in, const int* in_sizes, int n_in,
                              void* d_out, int out_size, void* d_ws, size_t ws_size,
                              hipStream_t stream) {
    (void)in_sizes; (void)n_in; (void)out_size; (void)ws_size;
    const float*         xx   = (const float*)d_in[0];
    const float*         kern = (const float*)d_in[1];
    const float*         bias = (const float*)d_in[2];
    const unsigned char* mask = (const unsigned char*)d_in[3];

    float* part  = (float*)d_ws;                       // NCH*B*D floats (2 MB)
    float* wsums = part + (size_t)NCH * B_ * D_;       // NCH*B floats

    dim3 grid(B_, NCH);                                // 1024 blocks
    attn_pool_stage1<<<grid, 256, 0, stream>>>(xx, kern, bias, mask, part, wsums);
    attn_finalize<<<B_, D_, 0, stream>>>(part, wsums, (float*)d_out);
}